// RandomParameterizedQuantumConvolutionLayer_58652073394702
// MI455X (gfx1250) — compile-verified
//
#include <hip/hip_runtime.h>
#include <hip/hip_bf16.h>
#include <stdint.h>
#include <string.h>

// ---------------------------------------------------------------------------
// Problem constants (shapes fixed by the reference: x=(512,1,64,64), NQUBIT=4,
// NUM_CIRCUITS=3, SEED=1024).
// ---------------------------------------------------------------------------
#define NQ      4
#define NCIRC   3
#define NGATES  15             // per circuit: 3 blocks * (4 single + 1 double)
#define WMAT_F  (NCIRC*32*32)  // 3 x 32x32 real-expanded unitaries in d_ws

struct GateSpec { int type; int q0; int q1; int off; };
struct CircuitDesc { GateSpec g[NCIRC * NGATES]; };

// ---------------------------------------------------------------------------
// Host-side CPython-compatible MT19937 (random.Random(1024)) to reconstruct
// the gate list exactly as build_circuits() does.
// ---------------------------------------------------------------------------
namespace pyrng {
struct MT {
  uint32_t mt[624]; int mti;
  void init_genrand(uint32_t s) {
    mt[0] = s;
    for (mti = 1; mti < 624; mti++)
      mt[mti] = 1812433253u * (mt[mti-1] ^ (mt[mti-1] >> 30)) + (uint32_t)mti;
  }
  void init_by_array(const uint32_t* key, int klen) {
    init_genrand(19650218u);
    int i = 1, j = 0;
    int k = (624 > klen) ? 624 : klen;
    for (; k; k--) {
      mt[i] = (mt[i] ^ ((mt[i-1] ^ (mt[i-1] >> 30)) * 1664525u)) + key[j] + (uint32_t)j;
      i++; j++;
      if (i >= 624) { mt[0] = mt[623]; i = 1; }
      if (j >= klen) j = 0;
    }
    for (k = 623; k; k--) {
      mt[i] = (mt[i] ^ ((mt[i-1] ^ (mt[i-1] >> 30)) * 1566083941u)) - (uint32_t)i;
      i++;
      if (i >= 624) { mt[0] = mt[623]; i = 1; }
    }
    mt[0] = 0x80000000u;
    mti = 624;
  }
  uint32_t genrand() {
    const uint32_t UP = 0x80000000u, LO = 0x7fffffffu;
    static const uint32_t mag01[2] = {0u, 0x9908b0dfu};
    if (mti >= 624) {
      int kk;
      for (kk = 0; kk < 624-397; kk++) {
        uint32_t y = (mt[kk] & UP) | (mt[kk+1] & LO);
        mt[kk] = mt[kk+397] ^ (y >> 1) ^ mag01[y & 1];
      }
      for (; kk < 623; kk++) {
        uint32_t y = (mt[kk] & UP) | (mt[kk+1] & LO);
        mt[kk] = mt[kk + (397-624)] ^ (y >> 1) ^ mag01[y & 1];
      }
      uint32_t y = (mt[623] & UP) | (mt[0] & LO);
      mt[623] = mt[396] ^ (y >> 1) ^ mag01[y & 1];
      mti = 0;
    }
    uint32_t y = mt[mti++];
    y ^= y >> 11; y ^= (y << 7) & 0x9d2c5680u; y ^= (y << 15) & 0xefc60000u; y ^= y >> 18;
    return y;
  }
  uint32_t getrandbits(int k) { return genrand() >> (32 - k); }
  uint32_t randbelow(uint32_t n) {         // CPython _randbelow_with_getrandbits
    int k = 32 - __builtin_clz(n);
    uint32_t r = getrandbits(k);
    while (r >= n) r = getrandbits(k);
    return r;
  }
};
} // namespace pyrng

// single types 0..6: rx ry rz s t p u3 ; double types 16+0..9: rxx ryy rzz swap cnot cp ch cu ct cz
static void build_circuits_host(CircuitDesc& cd) {
  pyrng::MT mt; uint32_t key = 1024u; mt.init_by_array(&key, 1);
  static const int pc1[7]  = {1,1,1,0,0,1,3};
  static const int pc2[10] = {1,1,1,0,0,1,0,3,0,0};
  int off = 0, gidx = 0;
  for (int c = 0; c < NCIRC; c++) {
    for (int blk = 0; blk < 3; blk++) {
      for (int i = 0; i < NQ; i++) {
        int g = (int)mt.randbelow(7);                  // rng.choice(SINGLE)
        cd.g[gidx++] = {g, i, 0, off};
        off += pc1[g];
      }
      int pool[3] = {0,1,2};                           // rng.sample(range(0,3),2)
      int j0 = (int)mt.randbelow(3); int cq = pool[j0]; pool[j0] = pool[2];
      int j1 = (int)mt.randbelow(2); int tq = pool[j1];
      int g = (int)mt.randbelow(10);                   // rng.choice(DOUBLE)
      cd.g[gidx++] = {16 + g, cq, tq, off};
      off += pc2[g];
    }
  }
}

// W storage layout (k-pair interleaved so a B fragment is one ds_load_b64):
//   element W[k][n]  ->  ws[ci*1024 + (k>>1)*64 + n*2 + (k&1)]
__host__ __device__ static inline int widx(int k, int n) {
  return ((k >> 1) * 64) + (n * 2) + (k & 1);
}

// ---------------------------------------------------------------------------
// Setup kernel: one thread per (circuit, basis column). Propagates e_col
// through the 15 gates -> column of the 16x16 complex unitary U, then writes
// the real-expanded 32x32 GEMM matrix W with signs baked in:
//   [Or|Oi] = [Sr|Si] * W,  W = [[Ur^T, Ui^T], [-Ui^T, Ur^T]]
// ---------------------------------------------------------------------------
__device__ static void build1(int tp, const float* p, float mr[2][2], float mi[2][2]) {
  for (int a = 0; a < 2; a++) for (int b = 0; b < 2; b++) { mr[a][b] = 0.f; mi[a][b] = 0.f; }
  const float R2 = 0.70710678118654752f;
  switch (tp) {
    case 0: { float h = 0.5f*p[0], c = cosf(h), s = sinf(h);                 // rx
      mr[0][0]=c; mr[1][1]=c; mi[0][1]=-s; mi[1][0]=-s; } break;
    case 1: { float h = 0.5f*p[0], c = cosf(h), s = sinf(h);                 // ry
      mr[0][0]=c; mr[0][1]=-s; mr[1][0]=s; mr[1][1]=c; } break;
    case 2: { float h = 0.5f*p[0], c = cosf(h), s = sinf(h);                 // rz
      mr[0][0]=c; mi[0][0]=-s; mr[1][1]=c; mi[1][1]=s; } break;
    case 3: mr[0][0]=1.f; mi[1][1]=1.f; break;                               // s
    case 4: mr[0][0]=1.f; mr[1][1]=R2; mi[1][1]=R2; break;                   // t
    case 5: mr[0][0]=1.f; mr[1][1]=cosf(p[0]); mi[1][1]=sinf(p[0]); break;   // p
    case 6: { float c = cosf(0.5f*p[0]), s = sinf(0.5f*p[0]);                // u3
      float cp = cosf(p[1]), sp = sinf(p[1]), cl = cosf(p[2]), sl = sinf(p[2]);
      mr[0][0]=c;
      mr[0][1]=-cl*s; mi[0][1]=-sl*s;
      mr[1][0]= cp*s; mi[1][0]= sp*s;
      mr[1][1]=(cp*cl - sp*sl)*c; mi[1][1]=(cp*sl + sp*cl)*c; } break;
  }
}

__device__ static void build2(int tp, const float* p, float mr[4][4], float mi[4][4]) {
  for (int a = 0; a < 4; a++) for (int b = 0; b < 4; b++) { mr[a][b] = 0.f; mi[a][b] = 0.f; }
  const float R2 = 0.70710678118654752f;
  switch (tp) {
    case 0: { float h = 0.5f*p[0], c = cosf(h), s = sinf(h);                 // rxx
      for (int d = 0; d < 4; d++) mr[d][d] = c;
      mi[0][3]=-s; mi[1][2]=-s; mi[2][1]=-s; mi[3][0]=-s; } break;
    case 1: { float h = 0.5f*p[0], c = cosf(h), s = sinf(h);                 // ryy
      for (int d = 0; d < 4; d++) mr[d][d] = c;
      mi[0][3]= s; mi[1][2]=-s; mi[2][1]=-s; mi[3][0]= s; } break;
    case 2: { float h = 0.5f*p[0], c = cosf(h), s = sinf(h);                 // rzz
      mr[0][0]=c; mi[0][0]=-s; mr[1][1]=c; mi[1][1]=s;
      mr[2][2]=c; mi[2][2]= s; mr[3][3]=c; mi[3][3]=-s; } break;
    case 3: mr[0][0]=1.f; mr[1][2]=1.f; mr[2][1]=1.f; mr[3][3]=1.f; break;   // swap
    case 4: mr[0][0]=1.f; mr[1][1]=1.f; mr[2][3]=1.f; mr[3][2]=1.f; break;   // cnot
    case 5: mr[0][0]=1.f; mr[1][1]=1.f; mr[2][2]=1.f;                        // cp
      mr[3][3]=cosf(p[0]); mi[3][3]=sinf(p[0]); break;
    case 6: mr[0][0]=1.f; mr[1][1]=1.f;                                      // ch
      mr[2][2]=R2; mr[2][3]=R2; mr[3][2]=R2; mr[3][3]=-R2; break;
    case 7: { mr[0][0]=1.f; mr[1][1]=1.f;                                    // cu
      float u3r[2][2], u3i[2][2];
      build1(6, p, u3r, u3i);
      for (int a = 0; a < 2; a++) for (int b = 0; b < 2; b++) {
        mr[2+a][2+b] = u3r[a][b]; mi[2+a][2+b] = u3i[a][b]; } } break;
    case 8: mr[0][0]=1.f; mr[1][1]=1.f; mr[2][2]=1.f;                        // ct
      mr[3][3]=R2; mi[3][3]=R2; break;
    case 9: mr[0][0]=1.f; mr[1][1]=1.f; mr[2][2]=1.f; mr[3][3]=-1.f; break;  // cz
  }
}

__global__ void qconv_setup(const float* __restrict__ params, float* __restrict__ W,
                            CircuitDesc cd) {
  int t = threadIdx.x;
  if (t >= NCIRC * 16) return;
  int ci = t >> 4, colb = t & 15;
  float sr[16], si[16];
  for (int j = 0; j < 16; j++) { sr[j] = (j == colb) ? 1.f : 0.f; si[j] = 0.f; }

  for (int gi = 0; gi < NGATES; gi++) {
    GateSpec g = cd.g[ci*NGATES + gi];
    if (g.type < 16) {
      float mr[2][2], mi[2][2];
      build1(g.type, params + g.off, mr, mi);
      int bit = 1 << (3 - g.q0);                     // qubit q -> bit (3-q)
      for (int j = 0; j < 16; j++) {
        if (j & bit) continue;
        int j1 = j | bit;
        float a0r = sr[j], a0i = si[j], a1r = sr[j1], a1i = si[j1];
        sr[j]  = mr[0][0]*a0r - mi[0][0]*a0i + mr[0][1]*a1r - mi[0][1]*a1i;
        si[j]  = mr[0][0]*a0i + mi[0][0]*a0r + mr[0][1]*a1i + mi[0][1]*a1r;
        sr[j1] = mr[1][0]*a0r - mi[1][0]*a0i + mr[1][1]*a1r - mi[1][1]*a1i;
        si[j1] = mr[1][0]*a0i + mi[1][0]*a0r + mr[1][1]*a1i + mi[1][1]*a1r;
      }
    } else {
      float mr[4][4], mi[4][4];
      build2(g.type - 16, params + g.off, mr, mi);
      int b0 = 1 << (3 - g.q0), b1 = 1 << (3 - g.q1);  // control, target
      for (int j = 0; j < 16; j++) {
        if (j & (b0 | b1)) continue;
        int idx[4] = { j, j | b1, j | b0, j | b0 | b1 };  // x = bc*2 + bt
        float ar[4], ai[4];
        for (int y = 0; y < 4; y++) { ar[y] = sr[idx[y]]; ai[y] = si[idx[y]]; }
        for (int xo = 0; xo < 4; xo++) {
          float nr = 0.f, ni = 0.f;
          for (int y = 0; y < 4; y++) {
            nr += mr[xo][y]*ar[y] - mi[xo][y]*ai[y];
            ni += mr[xo][y]*ai[y] + mi[xo][y]*ar[y];
          }
          sr[idx[xo]] = nr; si[idx[xo]] = ni;
        }
      }
    }
  }
  // state after gates = U[:, colb];  W rows k=colb and k=16+colb, interleaved
  float* Wc = W + ci * 1024;
  for (int n = 0; n < 16; n++) {
    Wc[widx(colb,      n)]      =  sr[n];   // Ur^T
    Wc[widx(colb,      16 + n)] =  si[n];   // Ui^T
    Wc[widx(16 + colb, n)]      = -si[n];   // -Ui^T
    Wc[widx(16 + colb, 16 + n)] =  sr[n];   // Ur^T
  }
}

// ---------------------------------------------------------------------------
// Main kernel. 256 threads = 8 wave32; each wave owns a 16-row tile.
// Per circuit: O(16x32) = X(16x32) * W(32x32) via 16x v_wmma_f32_16x16x4_f32.
// A layout (ISA 16x4 f32): lane holds M=lane&15, lane-half picks K {0,1}/{2,3}.
// B fragment for a lane = W[k:k+1][n], contiguous 8B in the interleaved
// layout -> single ds_load_b64 into an even-aligned VGPR pair.
// ---------------------------------------------------------------------------
typedef __attribute__((ext_vector_type(2))) float v2f;
typedef __attribute__((ext_vector_type(8))) float v8f;

__global__ __launch_bounds__(256) void qconv_main(const float* __restrict__ x,
                                                  const float* __restrict__ W,
                                                  float* __restrict__ out,
                                                  int nrows) {
  __shared__ float lds[WMAT_F];
  int tid = threadIdx.x;
  for (int i = tid; i < WMAT_F; i += 256) lds[i] = W[i];
  __syncthreads();

  int wave = tid >> 5, lane = tid & 31;
  int tile = blockIdx.x * 8 + wave;
  int row0 = tile * 16;
  if (row0 >= nrows) return;                   // wave-uniform exit

  int m   = lane & 15;                         // my M row within the tile
  int hi4 = lane >> 4;                         // lane-half -> K-offset select
  int row = row0 + m;

  // Gather the 2x2 patch for this row: x is (512,1,64,64); 1024 rows/batch.
  int b  = row >> 10;
  int r  = row & 1023;
  int f0 = r << 2;
  int h2 = f0 >> 7;
  int w0 = (f0 & 127) >> 1;
  const float* xb = x + b*4096 + (2*h2)*64 + w0;
  float ang[4] = { xb[0], xb[64], xb[1], xb[65] };   // qubits 0..3

  // Product state: amp(bas) = prod(bit? sin : cos) * (-i)^popcount(bas)
  float cs[4], sn[4];
  #pragma unroll
  for (int i = 0; i < 4; i++) { float h = 0.5f*ang[i]; cs[i] = __cosf(h); sn[i] = __sinf(h); }

  float Sr[16], Si[16];
  #pragma unroll
  for (int bas = 0; bas < 16; bas++) {
    float mag = 1.f; int pc = 0;
    #pragma unroll
    for (int i = 0; i < 4; i++) {
      int bit = (bas >> (3 - i)) & 1;
      mag *= bit ? sn[i] : cs[i];
      pc += bit;
    }
    int ph = pc & 3;
    Sr[bas] = (ph == 0) ? mag : (ph == 2) ? -mag : 0.f;
    Si[bas] = (ph == 1) ? -mag : (ph == 3) ? mag : 0.f;
  }

  // Per-lane A values: Xv[2k+e] = X[m][4k + 2*hi4 + e],  X = [Sr | Si]
  #define XCOL(c) ((c) < 16 ? Sr[(c)] : Si[(c) - 16])
  float Xv[16];
  #pragma unroll
  for (int kk = 0; kk < 8; kk++) {
    float lo0 = XCOL(4*kk + 0), lo1 = XCOL(4*kk + 1);
    float up0 = XCOL(4*kk + 2), up1 = XCOL(4*kk + 3);
    Xv[2*kk]   = hi4 ? up0 : lo0;
    Xv[2*kk+1] = hi4 ? up1 : lo1;
  }
  #undef XCOL

  for (int ci = 0; ci < NCIRC; ci++) {
    // B fragment pointers: k-pair index = 2*kk + hi4, column n (or 16+n).
    // Byte address = 4*(ci*1024 + (2*kk+hi4)*64 + n*2) -> 8B aligned.
    const v2f* B0 = (const v2f*)(lds + ci*1024 + hi4*64 + m*2);        // nt=0
    const v2f* B1 = (const v2f*)(lds + ci*1024 + hi4*64 + (16+m)*2);   // nt=1
    v8f acc0 = {}; v8f acc1 = {};
    #pragma unroll
    for (int kk = 0; kk < 8; kk++) {
      v2f a;  a.x = Xv[2*kk];  a.y = Xv[2*kk+1];
      v2f b0 = B0[kk*64];                      // ds_load_b64 (stride 512B)
      v2f b1 = B1[kk*64];
      acc0 = __builtin_amdgcn_wmma_f32_16x16x4_f32(false, a, false, b0,
                                                   (short)0, acc0, false, false);
      acc1 = __builtin_amdgcn_wmma_f32_16x16x4_f32(false, a, false, b1,
                                                   (short)0, acc1, false, false);
    }

    // probs: lane holds p[M = rr + 8*hi4][basis = lane&15] for rr=0..7
    float p[8];
    #pragma unroll
    for (int rr = 0; rr < 8; rr++) p[rr] = acc0[rr]*acc0[rr] + acc1[rr]*acc1[rr];

    // Walsh-Hadamard butterfly across the 16 basis lanes (masks < 16 keep
    // the shuffle within each lane-half). Lane j ends with y[j].
    #pragma unroll
    for (int st = 0; st < 4; st++) {
      int mask = 1 << st;
      #pragma unroll
      for (int rr = 0; rr < 8; rr++) {
        float o = __shfl_xor(p[rr], mask, 32);
        p[rr] = (lane & mask) ? (o - p[rr]) : (p[rr] + o);
      }
    }

    // e[q] = y[2^(3-q)] : lanes m = 8,4,2,1 hold q = 0,1,2,3
    int q = (m == 8) ? 0 : (m == 4) ? 1 : (m == 2) ? 2 : (m == 1) ? 3 : -1;
    if (q >= 0) {
      #pragma unroll
      for (int rr = 0; rr < 8; rr++) {
        int mm = rr + 8*hi4;
        out[(row0 + mm)*12 + ci*4 + q] = p[rr];
      }
    }
  }
}

// ---------------------------------------------------------------------------
// Launch
// ---------------------------------------------------------------------------
extern "C" void kernel_launch(void* const* d_in, const int* in_sizes, int n_in,
                              void* d_out, int out_size, void* d_ws, size_t ws_size,
                              hipStream_t stream) {
  const float* x      = (const float*)d_in[0];
  const float* params = (const float*)d_in[1];
  float* out          = (float*)d_out;
  float* W            = (float*)d_ws;          // 3*32*32 f32 = 12 KB

  CircuitDesc cd;
  build_circuits_host(cd);                     // deterministic, host-only

  qconv_setup<<<1, 64, 0, stream>>>(params, W, cd);

  int nrows  = in_sizes[0] / 4;                // 524288
  int blocks = (nrows + 127) / 128;            // 8 waves x 16 rows per block
  qconv_main<<<blocks, 256, 0, stream>>>(x, W, out, nrows);
}